// OneHotEncoding_63814624084361
// MI455X (gfx1250) — compile-verified
//
#include <hip/hip_runtime.h>

typedef __attribute__((ext_vector_type(2))) float v2f;
typedef __attribute__((ext_vector_type(4))) float v4f;
typedef __attribute__((ext_vector_type(8))) float v8f;

#define L_PTS   1000000   // 100*100*100 mesh points
#define N_RECV  128
#define CHUNK   1024      // points per block
#define NTHREADS 256      // 8 waves (wave32)

// ---------------------------------------------------------------------------
// Kernel 0: init the 128 per-receiver argmin slots to max (packed u64)
// ---------------------------------------------------------------------------
__global__ void nn_init_slots(unsigned long long* __restrict__ slots) {
    int t = threadIdx.x;
    if (t < N_RECV) slots[t] = ~0ULL;
}

__device__ __forceinline__ unsigned long long shfl_xor_u64(unsigned long long x, int m) {
    unsigned int lo = (unsigned int)x;
    unsigned int hi = (unsigned int)(x >> 32);
    lo = (unsigned int)__shfl_xor((int)lo, m, 32);
    hi = (unsigned int)__shfl_xor((int)hi, m, 32);
    return ((unsigned long long)hi << 32) | lo;
}

// ---------------------------------------------------------------------------
// Kernel 1: per block of 1024 points:
//   (a) write input_tensor rows [p | 0]  (one_hot column zeroed)
//   (b) WMMA metric m' = 8 + |p|^2 - 2 r.p  (argmin-equivalent to |r-p|^2,
//       and strictly positive so raw f32 bits order as unsigned ints),
//       per-lane running min via v_min_u64, butterfly reduce, atomicMin_u64
// ---------------------------------------------------------------------------
__global__ __launch_bounds__(NTHREADS) void nn_main(
        const float* __restrict__ pts,   // [L,3]
        const float* __restrict__ recv,  // [128,3]
        float* __restrict__ out,         // d_out: input_tensor at offset 0
        unsigned long long* __restrict__ slots) {
    const int lane = threadIdx.x & 31;
    const int wave = threadIdx.x >> 5;           // 0..7 -> receiver tile
    const int base = blockIdx.x * CHUNK;

    // ---- (a) stream the [L,4] tensor rows for this chunk (B128 stores) ----
    for (int i = threadIdx.x; i < CHUNK; i += NTHREADS) {
        int p = base + i;
        if (p < L_PTS) {
            float px = pts[p * 3 + 0];
            float py = pts[p * 3 + 1];
            float pz = pts[p * 3 + 2];
            v4f row = {px, py, pz, 0.0f};
            *(v4f*)(out + (size_t)p * 4) = row;
        }
    }

    // ---- (b) WMMA argmin ----
    const int half = lane >> 4;   // 0: holds K=0,1 ; 1: holds K=2,3
    const int col  = lane & 15;   // A: receiver row M ; B/C: point column N

    // A operand (16x4 f32): row M = (rx, ry, rz, 1)
    const int rid = wave * 16 + col;
    const float rx = recv[rid * 3 + 0];
    const float ry = recv[rid * 3 + 1];
    const float rz = recv[rid * 3 + 2];
    v2f a;
    a.x = half ? rz : rx;     // K=0 / K=2
    a.y = half ? 1.0f : ry;   // K=1 / K=3

    unsigned long long best[8];
#pragma unroll
    for (int v = 0; v < 8; ++v) best[v] = ~0ULL;

#pragma unroll 2
    for (int tb = 0; tb < CHUNK; tb += 16) {
        int pidx = base + tb + col;
        int pclamp = pidx < L_PTS ? pidx : (L_PTS - 1);  // dup loses tie on idx
        float px = pts[pclamp * 3 + 0];
        float py = pts[pclamp * 3 + 1];
        float pz = pts[pclamp * 3 + 2];
        // 8 + |p|^2 : bias seeds the FMA chain (free) and keeps metric > 0,
        // since 2 r.p < 6 for coords in [0,1). Positive f32 orders as u32.
        float n2b = fmaf(px, px, fmaf(py, py, fmaf(pz, pz, 8.0f)));

        // B operand (4x16 f32): column N = (-2px, -2py, -2pz, 8+|p|^2)
        v2f b;
        b.x = half ? (-2.0f * pz) : (-2.0f * px);  // K=0 / K=2
        b.y = half ? n2b : (-2.0f * py);           // K=1 / K=3

        v8f c = {0.f, 0.f, 0.f, 0.f, 0.f, 0.f, 0.f, 0.f};
        c = __builtin_amdgcn_wmma_f32_16x16x4_f32(
                /*neg_a=*/false, a, /*neg_b=*/false, b,
                /*c_mod=*/(short)0, c, /*reuse_a=*/false, /*reuse_b=*/false);

        unsigned long long idx = (unsigned int)pclamp;
#pragma unroll
        for (int v = 0; v < 8; ++v) {
            unsigned long long pk =
                ((unsigned long long)__float_as_uint(c[v]) << 32) | idx;
            best[v] = pk < best[v] ? pk : best[v];
        }
    }

    // butterfly min within each 16-lane half (columns of one receiver row)
#pragma unroll
    for (int off = 8; off >= 1; off >>= 1) {
#pragma unroll
        for (int v = 0; v < 8; ++v) {
            unsigned long long o = shfl_xor_u64(best[v], off);
            best[v] = o < best[v] ? o : best[v];
        }
    }

    if (col == 0) {  // lanes 0 (rows 0-7) and 16 (rows 8-15)
#pragma unroll
        for (int v = 0; v < 8; ++v) {
            int r = wave * 16 + v + (half ? 8 : 0);
            atomicMin(&slots[r], best[v]);
        }
    }
}

// ---------------------------------------------------------------------------
// Kernel 2: unpack winners -> min_index, closest_points, one_hot scatter
// d_out layout: input_tensor [L*4] | closest_points [128*3] | min_index [128]
// ---------------------------------------------------------------------------
__global__ void nn_finalize(const float* __restrict__ pts,
                            const unsigned long long* __restrict__ slots,
                            float* __restrict__ out) {
    int r = threadIdx.x;
    if (r < N_RECV) {
        unsigned long long pk = slots[r];
        int idx = (int)(unsigned int)(pk & 0xFFFFFFFFu);
        float* cp = out + (size_t)L_PTS * 4 + (size_t)r * 3;
        cp[0] = pts[idx * 3 + 0];
        cp[1] = pts[idx * 3 + 1];
        cp[2] = pts[idx * 3 + 2];
        out[(size_t)L_PTS * 4 + (size_t)N_RECV * 3 + r] = (float)idx;
        out[(size_t)idx * 4 + 3] = 1.0f;   // one_hot scatter
        if (r == 0) out[3] = 1.0f;         // reference's B>1 quirk: one_hot[0]=1
    }
}

extern "C" void kernel_launch(void* const* d_in, const int* in_sizes, int n_in,
                              void* d_out, int out_size, void* d_ws, size_t ws_size,
                              hipStream_t stream) {
    const float* pts  = (const float*)d_in[0];   // mesh_3D [100,100,100,3]
    const float* recv = (const float*)d_in[1];   // receiver_pos [128,3]
    float* out = (float*)d_out;
    unsigned long long* slots = (unsigned long long*)d_ws;  // 128 * 8B

    nn_init_slots<<<1, 128, 0, stream>>>(slots);
    int nblocks = (L_PTS + CHUNK - 1) / CHUNK;   // 977
    nn_main<<<nblocks, NTHREADS, 0, stream>>>(pts, recv, out, slots);
    nn_finalize<<<1, 128, 0, stream>>>(pts, slots, out);
}